// GraphSAGE_72361609003660
// MI455X (gfx1250) — compile-verified
//
#include <hip/hip_runtime.h>

#define N_NODES 100000
#define N_EDGES 1250000
#define IN_C 64
#define HID_C 64
#define OUT_C 32

typedef __attribute__((ext_vector_type(2))) float v2f;
typedef __attribute__((ext_vector_type(8))) float v8f;

// Non-returning device-scope f32 atomic add: guaranteed global_atomic_add_f32
// (STOREcnt-tracked, no CAS-loop fallback). DEV scope is required since waves
// from many WGPs hit the same agg rows; default scope would be WGP-coherent only.
__device__ __forceinline__ void atomic_add_f32(float* p, float v) {
    asm volatile("global_atomic_add_f32 %0, %1, off scope:SCOPE_DEV"
                 :
                 : "v"(p), "v"(v)
                 : "memory");
}

// ---------------------------------------------------------------------------
// Zero-fill
// ---------------------------------------------------------------------------
__global__ __launch_bounds__(256) void zero_kernel(float* __restrict__ p, long long n) {
    long long i = (long long)blockIdx.x * blockDim.x + threadIdx.x;
    if (i < n) p[i] = 0.0f;
}

// ---------------------------------------------------------------------------
// Degree count: deg[dst] += 1 per edge
// ---------------------------------------------------------------------------
__global__ __launch_bounds__(256) void deg_kernel(const long long* __restrict__ ei,
                                                  float* __restrict__ deg) {
    int e = blockIdx.x * blockDim.x + threadIdx.x;
    if (e >= N_EDGES) return;
    int dst = (int)ei[N_EDGES + e];
    atomic_add_f32(&deg[dst], 1.0f);
}

// ---------------------------------------------------------------------------
// Edge scatter: agg[dst][ch] += feat[src][ch].
// One thread per (edge, 4 channels): 16 consecutive threads cover one edge,
// so the float4 gather is a fully-coalesced 256B burst per edge and the
// atomics land on 4 consecutive floats per lane. Index traffic is 16B per
// 4 channels (4x less than per-channel threading). agg (25.6MB) fits in the
// 192MB L2 -> atomics resolve in L2.
// ---------------------------------------------------------------------------
__global__ __launch_bounds__(256) void scatter_kernel(const float* __restrict__ feat,
                                                      const long long* __restrict__ ei,
                                                      float* __restrict__ agg,
                                                      int total) {   // N_EDGES * 16
    int idx = blockIdx.x * blockDim.x + threadIdx.x;
    if (idx >= total) return;
    int e  = idx >> 4;
    int c4 = (idx & 15) << 2;
    int src = (int)ei[e];
    int dst = (int)ei[N_EDGES + e];
    const float4 v = *(const float4*)(feat + (size_t)src * 64 + c4);  // global_load_b128
    float* q = agg + (size_t)dst * 64 + c4;
    atomic_add_f32(q + 0, v.x);
    atomic_add_f32(q + 1, v.y);
    atomic_add_f32(q + 2, v.z);
    atomic_add_f32(q + 3, v.w);
}

// ---------------------------------------------------------------------------
// Fused WMMA GEMM tile kernel (fp32 V_WMMA_F32_16X16X4_F32).
//   out[16 x NTILES*16] = [HAS_AGG] (agg/deg) @ Wl^T  +  self @ Wr^T  + bias
//   optional ReLU.  One wave32 per 16-row M tile; K = 64.
//
// Fragment layouts (ISA 7.12.2, 32-bit):
//   A 16x4 : lane l holds A[l&15][k + 2*(l>>4)], A[l&15][k + 2*(l>>4) + 1]
//   B 4x16 : lane l holds B[k + 2*(l>>4)][l&15], B[k + 2*(l>>4) + 1][l&15]
//            with B = W^T  =>  load W[(l&15)][kk], W[(l&15)][kk+1] (b64 load)
//   C/D    : vgpr i, lane l -> D[i + 8*(l>>4)][l&15]
// ---------------------------------------------------------------------------
template <int NTILES, bool HAS_AGG, bool RELU>
__global__ __launch_bounds__(256) void sage_gemm_kernel(
    const float* __restrict__ self,   // [N, 64]
    const float* __restrict__ agg,    // [N, 64]  (HAS_AGG)
    const float* __restrict__ deg,    // [N]      (HAS_AGG)
    const float* __restrict__ Wl,     // [NTILES*16, 64] (HAS_AGG)
    const float* __restrict__ Wr,     // [NTILES*16, 64]
    const float* __restrict__ bias,   // [NTILES*16]
    float* __restrict__ out,          // [N, NTILES*16]
    int n_tiles_m) {
    const int wave = threadIdx.x >> 5;
    const int tile = blockIdx.x * 8 + wave;
    if (tile >= n_tiles_m) return;             // wave-uniform: EXEC stays all-1s for WMMA

    const int lane = threadIdx.x & 31;
    const int hf   = lane >> 4;                // 0 / 1 : K-half selector
    const int m    = lane & 15;                // row (A) / col (B,D) index
    const int row0 = tile * 16;
    const int OUTW = NTILES * 16;

    float scale = 0.0f;
    if constexpr (HAS_AGG) scale = 1.0f / fmaxf(deg[row0 + m], 1.0f);

    // C initialized with bias: D[row][col] += b[col], col = m (row-invariant per lane)
    v8f acc[NTILES];
#pragma unroll
    for (int j = 0; j < NTILES; ++j) {
        float bv = bias[j * 16 + m];
#pragma unroll
        for (int i = 0; i < 8; ++i) acc[j][i] = bv;
    }

    const float* selfrow = self + (size_t)(row0 + m) * 64;
    const float* aggrow  = HAS_AGG ? (agg + (size_t)(row0 + m) * 64) : nullptr;

#pragma unroll
    for (int k = 0; k < 64; k += 4) {
        const int kk = k + hf * 2;
        v2f a_r = *(const v2f*)(selfrow + kk);
        v2f a_l = {0.0f, 0.0f};
        if constexpr (HAS_AGG) {
            v2f t = *(const v2f*)(aggrow + kk);
            a_l.x = t.x * scale;
            a_l.y = t.y * scale;
        }
#pragma unroll
        for (int j = 0; j < NTILES; ++j) {
            v2f b_r = *(const v2f*)(Wr + (size_t)(j * 16 + m) * 64 + kk);
            acc[j] = __builtin_amdgcn_wmma_f32_16x16x4_f32(
                false, a_r, false, b_r, (short)0, acc[j], false, false);
            if constexpr (HAS_AGG) {
                v2f b_l = *(const v2f*)(Wl + (size_t)(j * 16 + m) * 64 + kk);
                acc[j] = __builtin_amdgcn_wmma_f32_16x16x4_f32(
                    false, a_l, false, b_l, (short)0, acc[j], false, false);
            }
        }
    }

#pragma unroll
    for (int j = 0; j < NTILES; ++j) {
#pragma unroll
        for (int i = 0; i < 8; ++i) {
            int row = row0 + i + hf * 8;
            float v = acc[j][i];
            if constexpr (RELU) v = fmaxf(v, 0.0f);
            out[(size_t)row * OUTW + j * 16 + m] = v;
        }
    }
}

// ---------------------------------------------------------------------------
// Launch: zero -> deg -> scatter(x) -> L1 gemm -> zero -> scatter(h1)
//         -> L2 gemm (writes h into d_out) -> classifier gemm (logits)
// ---------------------------------------------------------------------------
extern "C" void kernel_launch(void* const* d_in, const int* in_sizes, int n_in,
                              void* d_out, int out_size, void* d_ws, size_t ws_size,
                              hipStream_t stream) {
    const float*     x   = (const float*)d_in[0];
    const long long* ei  = (const long long*)d_in[1];
    const float*     W1l = (const float*)d_in[2];
    const float*     b1l = (const float*)d_in[3];
    const float*     W1r = (const float*)d_in[4];
    const float*     W2l = (const float*)d_in[5];
    const float*     b2l = (const float*)d_in[6];
    const float*     W2r = (const float*)d_in[7];
    const float*     Wc  = (const float*)d_in[8];
    const float*     bc  = (const float*)d_in[9];

    float* agg = (float*)d_ws;                    // [N, 64]
    float* deg = agg + (size_t)N_NODES * 64;      // [N]
    float* h1  = deg + N_NODES;                   // [N, 64]

    float* logits = (float*)d_out;                        // [N, 32]
    float* h2     = logits + (size_t)N_NODES * OUT_C;     // [N, 64] (second tuple output)

    const int TPB = 256;
    const long long aggN = (long long)N_NODES * 64;
    const int scThreads  = N_EDGES * 16;          // 20M: one thread per (edge, 4ch)
    const int mtiles  = N_NODES / 16;             // 6250 (exact)
    const int gblocks = (mtiles + 7) / 8;         // 8 waves (tiles) per block

    // --- Layer 1 aggregation ---
    zero_kernel<<<(int)((aggN + N_NODES + TPB - 1) / TPB), TPB, 0, stream>>>(agg, aggN + N_NODES); // agg+deg contiguous
    deg_kernel<<<(N_EDGES + TPB - 1) / TPB, TPB, 0, stream>>>(ei, deg);
    scatter_kernel<<<(scThreads + TPB - 1) / TPB, TPB, 0, stream>>>(x, ei, agg, scThreads);

    // h1 = relu(mean @ W1_l^T + b1_l + x @ W1_r^T)
    sage_gemm_kernel<4, true, true><<<gblocks, 256, 0, stream>>>(
        x, agg, deg, W1l, W1r, b1l, h1, mtiles);

    // --- Layer 2 aggregation ---
    zero_kernel<<<(int)((aggN + TPB - 1) / TPB), TPB, 0, stream>>>(agg, aggN);
    scatter_kernel<<<(scThreads + TPB - 1) / TPB, TPB, 0, stream>>>(h1, ei, agg, scThreads);

    // h2 = relu(mean @ W2_l^T + b2_l + h1 @ W2_r^T) -> directly into d_out slice
    sage_gemm_kernel<4, true, true><<<gblocks, 256, 0, stream>>>(
        h1, agg, deg, W2l, W2r, b2l, h2, mtiles);

    // logits = h2 @ Wc^T + bc
    sage_gemm_kernel<2, false, false><<<gblocks, 256, 0, stream>>>(
        h2, nullptr, nullptr, nullptr, Wc, bc, logits, mtiles);
}